// GAE_6201932775427
// MI455X (gfx1250) — compile-verified
//
#include <hip/hip_runtime.h>
#include <hip/hip_bf16.h>

// ---------------------------------------------------------------------------
// 2-layer GCN forward for MI455X (gfx1250, wave32).
//   dinv = rsqrt(1 + indeg)
//   g1   = dinv * (x @ W1)          (fp32 WMMA 16x16x4, dinv fused in epilogue)
//   agg1 = dinv[i]*g1[i] + sum_e dinv[d]*g1[s]   (L2-resident f32 atomics)
//   h    = relu(agg1 + b1)
//   g2   = dinv * (h @ W2)
//   out  = relu(dinv[i]*g2[i] + scatter + b2)
// ---------------------------------------------------------------------------

typedef float v2f __attribute__((ext_vector_type(2)));
typedef float v8f __attribute__((ext_vector_type(8)));

#if __has_builtin(__builtin_amdgcn_global_load_async_to_lds_b128) && \
    __has_builtin(__builtin_amdgcn_s_wait_asynccnt)
#define HAS_ASYNC_LDS 1
typedef int v4i_vs __attribute__((vector_size(16)));
typedef void    __attribute__((address_space(1))) gvoid;
typedef void    __attribute__((address_space(3))) lvoid;
typedef v4i_vs  __attribute__((address_space(1))) gvec_t;
typedef v4i_vs  __attribute__((address_space(3))) lvec_t;
#else
#define HAS_ASYNC_LDS 0
#endif

// ---------------- degree / normalization ----------------------------------

__global__ void k_deg_init(float* __restrict__ deg, int N) {
    int i = blockIdx.x * blockDim.x + threadIdx.x;
    if (i < N) deg[i] = 1.0f;  // self-loop
}

__global__ void k_deg_count(const int* __restrict__ dst, float* __restrict__ deg, int E) {
    int e = blockIdx.x * blockDim.x + threadIdx.x;
    if (e < E) atomicAdd(&deg[dst[e]], 1.0f);
}

__global__ void k_dinv(float* __restrict__ deg, int N) {
    int i = blockIdx.x * blockDim.x + threadIdx.x;
    if (i < N) deg[i] = __frsqrt_rn(deg[i]);   // deg >= 1 always
}

// ---------------- fp32 WMMA GEMM: C = dinv[m] * (A[MxK] @ B[KxNc]) --------
// Block = W waves; wave w owns a 16x64 tile (4 accumulators -> 4 independent
// WMMA chains fed by one ds_load_b64 A-fragment per k-step). The 16xK A strip
// is staged into LDS via GLOBAL_LOAD_ASYNC_TO_LDS_B128 (ASYNCcnt) with a
// 260-float pitch (conflict-free fragment reads).

#define LDS_PITCH 260

__global__ __launch_bounds__(128) void k_gemm_wmma(
    const float* __restrict__ A, const float* __restrict__ B,
    const float* __restrict__ dinv, float* __restrict__ C,
    int M, int K, int Nc, int W) {

    __shared__ __align__(16) float As[16 * LDS_PITCH];

    const int tid  = threadIdx.x;
    const int nthr = blockDim.x;
    const int row0 = blockIdx.x * 16;
    const int K4   = K >> 2;

    // stage the 16 x K strip of A into LDS (direct-to-LDS async b128)
    for (int i = tid; i < 16 * K4; i += nthr) {
        int r  = i / K4;
        int c4 = i - r * K4;
        int rr = row0 + r; if (rr >= M) rr = M - 1;
        const float* gp = A + (size_t)rr * K + (size_t)c4 * 4;
        float*       lp = &As[r * LDS_PITCH + c4 * 4];
#if HAS_ASYNC_LDS
        __builtin_amdgcn_global_load_async_to_lds_b128(
            (gvec_t*)(gvoid*)gp, (lvec_t*)(lvoid*)lp, 0, 0);
#else
        *(float4*)lp = *(const float4*)gp;
#endif
    }
#if HAS_ASYNC_LDS
    __builtin_amdgcn_s_wait_asynccnt(0);
#endif
    __syncthreads();

    const int wave  = tid >> 5;
    const int lane  = tid & 31;
    const int half  = lane >> 4;       // 0: lanes 0-15, 1: lanes 16-31
    const int mr    = lane & 15;       // A row / B,C column within tile
    const int nbase = (blockIdx.y * W + wave) * 64;
    if (nbase >= Nc) return;           // uniform per wave

    v8f c0 = {0.f,0.f,0.f,0.f,0.f,0.f,0.f,0.f};
    v8f c1 = c0, c2 = c0, c3 = c0;

#pragma unroll 2
    for (int k = 0; k < K; k += 4) {
        const int kk = k + 2 * half;
        // A fragment: lane l -> A[l&15][kk], A[l&15][kk+1]
        v2f a = *(const v2f*)&As[mr * LDS_PITCH + kk];
        // B fragments for 4 column tiles: lane l -> B[kk][n], B[kk+1][n]
        const float* br0 = B + (size_t)kk * Nc + nbase + mr;
        const float* br1 = br0 + Nc;
        v2f b0 = { br0[0],  br1[0]  };
        v2f b1 = { br0[16], br1[16] };
        v2f b2 = { br0[32], br1[32] };
        v2f b3 = { br0[48], br1[48] };
        c0 = __builtin_amdgcn_wmma_f32_16x16x4_f32(false, a, false, b0, (short)0, c0, false, false);
        c1 = __builtin_amdgcn_wmma_f32_16x16x4_f32(false, a, false, b1, (short)0, c1, false, false);
        c2 = __builtin_amdgcn_wmma_f32_16x16x4_f32(false, a, false, b2, (short)0, c2, false, false);
        c3 = __builtin_amdgcn_wmma_f32_16x16x4_f32(false, a, false, b3, (short)0, c3, false, false);
    }

    // epilogue: C[m][n] = dinv[m] * acc   (C fragment row m = r + 8*half)
#pragma unroll
    for (int r = 0; r < 8; ++r) {
        int m = row0 + r + 8 * half;
        if (m < M) {
            float s = dinv[m];
            size_t base = (size_t)m * Nc + nbase + mr;
            C[base]      = s * c0[r];
            C[base + 16] = s * c1[r];
            C[base + 32] = s * c2[r];
            C[base + 48] = s * c3[r];
        }
    }
}

// ---------------- aggregation --------------------------------------------

// agg[i] = dinv[i] * g[i]   (self-loop term; fully initializes agg)
__global__ void k_selfloop(const float* __restrict__ dinv, const float* __restrict__ g,
                           float* __restrict__ agg, int N, int F4) {
    int t = blockIdx.x * blockDim.x + threadIdx.x;
    int i = t / F4;
    int j = t - i * F4;
    if (i >= N) return;
    float w = dinv[i];
    float4 v = ((const float4*)(g + (size_t)i * F4 * 4))[j];
    float4 o = make_float4(w * v.x, w * v.y, w * v.z, w * v.w);
    ((float4*)(agg + (size_t)i * F4 * 4))[j] = o;
}

// one edge handled by F4 consecutive threads: coalesced b128 gather of
// g[src], f32 atomic adds (L2-resident accumulator). norm = dinv[d] only,
// since dinv[s] was folded into g by the GEMM epilogue.
__global__ void k_scatter(const int* __restrict__ src, const int* __restrict__ dst,
                          const float* __restrict__ dinv, const float* __restrict__ g,
                          float* __restrict__ agg, int E, int F4) {
    int t = blockIdx.x * blockDim.x + threadIdx.x;
    int e = t / F4;
    int j = t - e * F4;
    if (e >= E) return;
    int s = src[e], d = dst[e];
    float w = dinv[d];
    float4 v = ((const float4*)(g + (size_t)s * F4 * 4))[j];
    float* p = agg + (size_t)d * F4 * 4 + (size_t)j * 4;
    atomicAdd(p + 0, w * v.x);
    atomicAdd(p + 1, w * v.y);
    atomicAdd(p + 2, w * v.z);
    atomicAdd(p + 3, w * v.w);
}

// out = relu(agg + bias)   (out may alias agg)
__global__ void k_bias_relu(const float* __restrict__ agg, const float* __restrict__ bias,
                            float* __restrict__ out, int N, int F4) {
    int t = blockIdx.x * blockDim.x + threadIdx.x;
    int i = t / F4;
    int j = t - i * F4;
    if (i >= N) return;
    float4 v = ((const float4*)(agg + (size_t)i * F4 * 4))[j];
    float4 b = ((const float4*)bias)[j];
    float4 o;
    o.x = fmaxf(v.x + b.x, 0.f);
    o.y = fmaxf(v.y + b.y, 0.f);
    o.z = fmaxf(v.z + b.z, 0.f);
    o.w = fmaxf(v.w + b.w, 0.f);
    ((float4*)(out + (size_t)i * F4 * 4))[j] = o;
}

// ---------------- driver ---------------------------------------------------

static inline int cdiv(long long a, int b) { return (int)((a + b - 1) / b); }

extern "C" void kernel_launch(void* const* d_in, const int* in_sizes, int n_in,
                              void* d_out, int out_size, void* d_ws, size_t ws_size,
                              hipStream_t stream) {
    const float* x  = (const float*)d_in[0];
    const int*   ei = (const int*)d_in[1];     // [2, E], row0 = src, row1 = dst
    const float* W1 = (const float*)d_in[2];
    const float* b1 = (const float*)d_in[3];
    const float* W2 = (const float*)d_in[4];
    const float* b2 = (const float*)d_in[5];

    const int HID = in_sizes[3];               // 256
    const int OUT = in_sizes[5];               // 128
    const int IN  = in_sizes[2] / HID;         // 256
    const int N   = in_sizes[0] / IN;          // 50000
    const int E   = in_sizes[1] / 2;           // 800000
    const int* src = ei;
    const int* dst = ei + E;

    float* ws   = (float*)d_ws;
    float* dinv = ws;
    size_t off  = ((size_t)N + 63) & ~(size_t)63;
    float* bufA = ws + off;                     // g1 [N x HID], later g2 [N x OUT]
    float* bufB = bufA + (size_t)N * HID;       // agg1 / h [N x HID]
    float* out  = (float*)d_out;                // agg2 / z [N x OUT]

    const int B = 256;

    // --- normalization -----------------------------------------------------
    k_deg_init <<<cdiv(N, B), B, 0, stream>>>(dinv, N);
    k_deg_count<<<cdiv(E, B), B, 0, stream>>>(dst, dinv, E);
    k_dinv     <<<cdiv(N, B), B, 0, stream>>>(dinv, N);

    // --- layer 1 -----------------------------------------------------------
    {
        const int W = (HID >= 256) ? 4 : HID / 64;     // waves per block
        dim3 g(cdiv(N, 16), HID / (W * 64));
        k_gemm_wmma<<<g, 32 * W, 0, stream>>>(x, W1, dinv, bufA, N, IN, HID, W);
        const int F4 = HID / 4;
        k_selfloop <<<cdiv((long long)N * F4, B), B, 0, stream>>>(dinv, bufA, bufB, N, F4);
        k_scatter  <<<cdiv((long long)E * F4, B), B, 0, stream>>>(src, dst, dinv, bufA, bufB, E, F4);
        k_bias_relu<<<cdiv((long long)N * F4, B), B, 0, stream>>>(bufB, b1, bufB, N, F4);
    }

    // --- layer 2 -----------------------------------------------------------
    {
        const int W = (OUT >= 256) ? 4 : OUT / 64;
        dim3 g(cdiv(N, 16), OUT / (W * 64));
        k_gemm_wmma<<<g, 32 * W, 0, stream>>>(bufB, W2, dinv, bufA, N, HID, OUT, W);
        const int F4 = OUT / 4;
        k_selfloop <<<cdiv((long long)N * F4, B), B, 0, stream>>>(dinv, bufA, out, N, F4);
        k_scatter  <<<cdiv((long long)E * F4, B), B, 0, stream>>>(src, dst, dinv, bufA, out, E, F4);
        k_bias_relu<<<cdiv((long long)N * F4, B), B, 0, stream>>>(out, b2, out, N, F4);
    }
}